// NetGIN_48404281426016
// MI455X (gfx1250) — compile-verified
//
#include <hip/hip_runtime.h>
#include <math.h>

typedef __attribute__((ext_vector_type(2))) float v2f;
typedef __attribute__((ext_vector_type(8))) float v8f;
typedef __attribute__((ext_vector_type(4))) unsigned v4u;
typedef __attribute__((ext_vector_type(8))) unsigned v8u;

#define N_NODES 50000
#define N_EDGES 800000
#define N_GRAPHS 64
#define DIM 64
#define BN_EPS 1e-5f

// ---------------- TDM: DMA a contiguous w x h f32 tile from global to LDS ----------------
// D# group0: [1:0]=count(1), [63:32]=lds byte addr, [120:64]=global addr, [127:126]=type(2)
// D# group1: data_size=4B (code 2), tensor_dim0=w, tensor_dim1=h, tile_dim0=w, tile_dim1=h,
//            tensor_dim0_stride=w, everything else 0 (no cluster mask / pad / iterate).
__device__ __forceinline__ void tdm_load_f32_2d(float* lds_dst, const float* gsrc,
                                                unsigned w, unsigned h) {
  unsigned lo = (unsigned)(unsigned long long)(uintptr_t)lds_dst;  // low 32 bits = LDS offset
  unsigned long long ga = (unsigned long long)(uintptr_t)gsrc;
  v4u g0;
  g0.x = 1u;                 // count = 1 valid descriptor
  g0.y = lo;                 // lds_addr
  g0.z = (unsigned)ga;       // global_addr[31:0]
  g0.w = (unsigned)((ga >> 32) & 0x1FFFFFFull) | 0x80000000u;  // addr[56:32] | type=2
  v8u g1;
  g1[0] = 0x20000u;              // data_size = 2 (4 bytes)
  g1[1] = (w & 0xFFFFu) << 16;   // tensor_dim0[15:0]
  g1[2] = (h & 0xFFFFu) << 16;   // tensor_dim1[15:0]
  g1[3] = (w & 0xFFFFu) << 16;   // tile_dim0
  g1[4] = (h & 0xFFFFu);         // tile_dim1 (tile_dim2 = 0)
  g1[5] = w;                     // tensor_dim0_stride[31:0]
  g1[6] = 0u;
  g1[7] = 0u;
  asm volatile("tensor_load_to_lds %0, %1" :: "s"(g0), "s"(g1) : "memory");
}

// ---------------- utility ----------------
__global__ void zero_f32(float* p, int n) {
  for (int i = blockIdx.x * blockDim.x + threadIdx.x; i < n; i += gridDim.x * blockDim.x)
    p[i] = 0.f;
}

// ---------------- conv1 (dim1 = 6, scalar) ----------------
__global__ void edge_conv1(const float* __restrict__ x, const float* __restrict__ ea,
                           const int* __restrict__ src, const int* __restrict__ dst,
                           const float* __restrict__ W1, const float* __restrict__ b1,
                           const float* __restrict__ W2, const float* __restrict__ b2,
                           float* __restrict__ agg) {
  int e = blockIdx.x * blockDim.x + threadIdx.x;
  if (e >= N_EDGES) return;
  float a0 = ea[e * 3 + 0], a1 = ea[e * 3 + 1], a2 = ea[e * 3 + 2];
  float t[6];
#pragma unroll
  for (int c = 0; c < 6; c++) {
    float v = b1[c] + a0 * W1[c] + a1 * W1[6 + c] + a2 * W1[12 + c];
    t[c] = v > 0.f ? v : 0.f;
  }
  int s = src[e], d = dst[e];
#pragma unroll
  for (int c = 0; c < 6; c++) {
    float v = b2[c];
#pragma unroll
    for (int k = 0; k < 6; k++) v += t[k] * W2[k * 6 + c];
    v += x[s * 6 + c];
    v = v > 0.f ? v : 0.f;
    atomicAdd(&agg[d * 64 + c], v);
  }
}

__global__ void node_conv1(const float* __restrict__ x, const float* __restrict__ agg,
                           const float* __restrict__ W1, const float* __restrict__ b1,
                           const float* __restrict__ W2, const float* __restrict__ b2,
                           const float* __restrict__ eps, float* __restrict__ y) {
  int n = blockIdx.x * blockDim.x + threadIdx.x;
  if (n >= N_NODES) return;
  float ep = 1.f + eps[0];
  float h[6], g[6];
#pragma unroll
  for (int c = 0; c < 6; c++) h[c] = ep * x[n * 6 + c] + agg[n * 64 + c];
#pragma unroll
  for (int c = 0; c < 6; c++) {
    float v = b1[c];
#pragma unroll
    for (int k = 0; k < 6; k++) v += h[k] * W1[k * 6 + c];
    g[c] = v > 0.f ? v : 0.f;
  }
  for (int j = 0; j < 64; j++) {
    float v = b2[j];
#pragma unroll
    for (int k = 0; k < 6; k++) v += g[k] * W2[k * 64 + j];
    y[n * 64 + j] = v > 0.f ? v : 0.f;
  }
}

// ---------------- WMMA edge kernel (conv2-4) ----------------
// Wave handles 16 edges: e = relu(ea@W1 + b1) @ W2 + b2 via V_WMMA_F32_16X16X4_F32,
// then msg = relu(x[src] + e) scattered with f32 atomics.
// W2 held in LDS in paired-row layout: one ds_load_b64 per B fragment.
__global__ __launch_bounds__(128) void edge_wmma(
    const float* __restrict__ x, const float* __restrict__ ea,
    const int* __restrict__ src, const int* __restrict__ dst,
    const float* __restrict__ W1, const float* __restrict__ b1,
    const float* __restrict__ W2, const float* __restrict__ b2,
    float* __restrict__ agg) {
  __shared__ float sW1[3 * 64];
  __shared__ float sB1[64];
  __shared__ float sB2[64];
  __shared__ v2f sW2p[32 * 65];  // pair p holds rows {2p, 2p+1}; stride 65 kills bank conflicts
  int tid = threadIdx.x;
  int wave = tid >> 5, lane = tid & 31;
  if (tid < 64) { sB1[tid] = b1[tid]; sB2[tid] = b2[tid]; }
  for (int idx = tid; idx < 32 * 64; idx += 128) {
    int p = idx >> 6, col = idx & 63;
    v2f t;
    t.x = W2[(2 * p) * 64 + col];
    t.y = W2[(2 * p + 1) * 64 + col];
    sW2p[p * 65 + col] = t;
  }
  if (wave == 0) {  // TDM DMA for the 3x64 bond-encoder weight
    tdm_load_f32_2d(sW1, W1, 64u, 3u);
    __builtin_amdgcn_s_wait_tensorcnt(0);
  }
  __syncthreads();

  int m = lane & 15, hl = lane >> 4;
  int eBase = (blockIdx.x * 4 + wave) * 16;  // 12500 blocks * 64 edges = 800000 exactly

  const float* eap = ea + (long)(eBase + m) * 3;
  float a0 = eap[0], a1 = eap[1], a2 = eap[2];

  // A fragments: t[m][4s+2h .. 4s+2h+1] (ISA 16x4 f32 A layout)
  v2f afrag[16];
#pragma unroll
  for (int s = 0; s < 16; s++) {
    int c0 = 4 * s + 2 * hl;
    float t0 = sB1[c0] + a0 * sW1[c0] + a1 * sW1[64 + c0] + a2 * sW1[128 + c0];
    float t1 = sB1[c0 + 1] + a0 * sW1[c0 + 1] + a1 * sW1[64 + c0 + 1] + a2 * sW1[128 + c0 + 1];
    afrag[s].x = t0 > 0.f ? t0 : 0.f;
    afrag[s].y = t1 > 0.f ? t1 : 0.f;
  }

  v8f acc[4];
#pragma unroll
  for (int nt = 0; nt < 4; nt++) {
    v8f c = {0.f, 0.f, 0.f, 0.f, 0.f, 0.f, 0.f, 0.f};
#pragma unroll
    for (int s = 0; s < 16; s++) {
      v2f bf = sW2p[(2 * s + hl) * 65 + nt * 16 + m];  // {W2[4s+2h][col], W2[4s+2h+1][col]}
      c = __builtin_amdgcn_wmma_f32_16x16x4_f32(false, afrag[s], false, bf,
                                                (short)0, c, false, false);
    }
    acc[nt] = c;
  }

  int se[8], de[8];
#pragma unroll
  for (int r = 0; r < 8; r++) {
    int e = eBase + r + 8 * hl;  // D layout: VGPR r -> row r + 8*half
    se[r] = src[e];
    de[r] = dst[e];
  }
#pragma unroll
  for (int nt = 0; nt < 4; nt++) {
    int col = nt * 16 + m;
    float bb = sB2[col];
#pragma unroll
    for (int r = 0; r < 8; r++) {
      float v = acc[nt][r] + bb + x[(long)se[r] * 64 + col];
      v = v > 0.f ? v : 0.f;
      atomicAdd(&agg[(long)de[r] * 64 + col], v);
    }
  }
}

// ---------------- WMMA node kernel (conv2-4) ----------------
// Wave handles 16 nodes: h=(1+eps)x+agg; y=relu(relu(h@W1+b1)@W2+b2).
__global__ __launch_bounds__(128) void node_wmma(
    const float* __restrict__ xin, const float* __restrict__ agg,
    const float* __restrict__ W1, const float* __restrict__ b1,
    const float* __restrict__ W2, const float* __restrict__ b2,
    const float* __restrict__ eps, float* __restrict__ y) {
  __shared__ v2f sW1p[32 * 65];
  __shared__ v2f sW2p[32 * 65];
  __shared__ float sB1[64], sB2[64];
  __shared__ float stage[4][16 * 68];  // pad to 68 -> conflict-free A-frag reads
  int tid = threadIdx.x;
  for (int idx = tid; idx < 32 * 64; idx += 128) {
    int p = idx >> 6, col = idx & 63;
    v2f t1, t2;
    t1.x = W1[(2 * p) * 64 + col];
    t1.y = W1[(2 * p + 1) * 64 + col];
    t2.x = W2[(2 * p) * 64 + col];
    t2.y = W2[(2 * p + 1) * 64 + col];
    sW1p[p * 65 + col] = t1;
    sW2p[p * 65 + col] = t2;
  }
  if (tid < 64) { sB1[tid] = b1[tid]; sB2[tid] = b2[tid]; }
  __syncthreads();

  int wave = tid >> 5, lane = tid & 31;
  int m = lane & 15, hl = lane >> 4;
  int tile = blockIdx.x * 4 + wave;
  bool act = tile < (N_NODES / 16);
  float ep = 1.f + eps[0];
  int base = tile * 16;
  float* st = &stage[wave][0];

  if (act) {
    for (int i = lane; i < 16 * 64; i += 32) {
      int r = i >> 6, c = i & 63;
      long g = (long)(base + r) * 64 + c;
      st[r * 68 + c] = ep * xin[g] + agg[g];
    }
  }
  __syncthreads();

  v8f acc[4];
  if (act) {
#pragma unroll
    for (int nt = 0; nt < 4; nt++) {
      v8f c = {0.f, 0.f, 0.f, 0.f, 0.f, 0.f, 0.f, 0.f};
#pragma unroll
      for (int s = 0; s < 16; s++) {
        int c0 = 4 * s + 2 * hl;
        v2f a = *(const v2f*)&st[m * 68 + c0];
        v2f bf = sW1p[(2 * s + hl) * 65 + nt * 16 + m];
        c = __builtin_amdgcn_wmma_f32_16x16x4_f32(false, a, false, bf, (short)0, c, false, false);
      }
      acc[nt] = c;
    }
  }
  __syncthreads();
  if (act) {
#pragma unroll
    for (int nt = 0; nt < 4; nt++) {
      int col = nt * 16 + m;
      float bb = sB1[col];
#pragma unroll
      for (int r = 0; r < 8; r++) {
        float v = acc[nt][r] + bb;
        st[(r + 8 * hl) * 68 + col] = v > 0.f ? v : 0.f;
      }
    }
  }
  __syncthreads();
  if (act) {
#pragma unroll
    for (int nt = 0; nt < 4; nt++) {
      v8f c = {0.f, 0.f, 0.f, 0.f, 0.f, 0.f, 0.f, 0.f};
#pragma unroll
      for (int s = 0; s < 16; s++) {
        int c0 = 4 * s + 2 * hl;
        v2f a = *(const v2f*)&st[m * 68 + c0];
        v2f bf = sW2p[(2 * s + hl) * 65 + nt * 16 + m];
        c = __builtin_amdgcn_wmma_f32_16x16x4_f32(false, a, false, bf, (short)0, c, false, false);
      }
      int col = nt * 16 + m;
      float bb = sB2[col];
#pragma unroll
      for (int r = 0; r < 8; r++) {
        float v = c[r] + bb;
        y[(long)(base + r + 8 * hl) * 64 + col] = v > 0.f ? v : 0.f;
      }
    }
  }
}

// ---------------- batch norm ----------------
__global__ void col_stats(const float* __restrict__ y, float* __restrict__ sums,
                          float* __restrict__ sumsq) {
  __shared__ float ls[8][64], lq[8][64];
  int c = threadIdx.x, ty = threadIdx.y;
  long row0 = (long)blockIdx.x * 512;
  float s = 0.f, q = 0.f;
  for (int rr = ty; rr < 512; rr += 8) {
    long r = row0 + rr;
    if (r < N_NODES) {
      float v = y[r * 64 + c];
      s += v; q += v * v;
    }
  }
  ls[ty][c] = s; lq[ty][c] = q;
  __syncthreads();
  if (ty == 0) {
#pragma unroll
    for (int k = 1; k < 8; k++) { s += ls[k][c]; q += lq[k][c]; }
    atomicAdd(&sums[c], s);
    atomicAdd(&sumsq[c], q);
  }
}

__global__ void bn_finalize(const float* sums, const float* sumsq, const float* g,
                            const float* b, float* scale, float* shift) {
  int c = threadIdx.x;
  float mean = sums[c] / (float)N_NODES;
  float var = sumsq[c] / (float)N_NODES - mean * mean;
  float sc = g[c] * rsqrtf(var + BN_EPS);
  scale[c] = sc;
  shift[c] = b[c] - mean * sc;
}

__global__ void bn_apply(const float* __restrict__ y, const float* __restrict__ scale,
                         const float* __restrict__ shift, float* __restrict__ xo) {
  int i = blockIdx.x * blockDim.x + threadIdx.x;  // exactly N_NODES*64 threads
  int c = i & 63;
  xo[i] = y[i] * scale[c] + shift[c];
}

// ---------------- pooling + head ----------------
__global__ void count_nodes(const int* __restrict__ batch, float* __restrict__ cnt) {
  int n = blockIdx.x * blockDim.x + threadIdx.x;
  if (n < N_NODES) atomicAdd(&cnt[batch[n]], 1.f);
}

__global__ void pool_sum(const float* __restrict__ x1, const float* __restrict__ x2,
                         const float* __restrict__ x3, const float* __restrict__ x4,
                         const int* __restrict__ batch, float* __restrict__ pool) {
  int i = blockIdx.x * blockDim.x + threadIdx.x;  // N_NODES*64 threads exactly
  int n = i >> 6, c = i & 63;
  int b = batch[n];
  atomicAdd(&pool[b * 256 + c], x1[i]);
  atomicAdd(&pool[b * 256 + 64 + c], x2[i]);
  atomicAdd(&pool[b * 256 + 128 + c], x3[i]);
  atomicAdd(&pool[b * 256 + 192 + c], x4[i]);
}

__global__ void pool_mean(const float* pool, const float* cnt, float* g) {
  int b = blockIdx.x, c = threadIdx.x;
  float n = cnt[b];
  n = n > 1.f ? n : 1.f;
  g[b * 256 + c] = pool[b * 256 + c] / n;
}

__global__ void head(const float* __restrict__ g,
                     const float* W1, const float* B1, const float* W2, const float* B2,
                     const float* W3, const float* B3, const float* W4, const float* B4,
                     float* __restrict__ out) {
  int gr = threadIdx.x;  // 64 graphs
  float h1[64], h2[64];
  for (int j = 0; j < 64; j++) {
    float v = B1[j];
    for (int k = 0; k < 256; k++) v += g[gr * 256 + k] * W1[k * 64 + j];
    h1[j] = v > 0.f ? v : 0.f;
  }
  for (int j = 0; j < 64; j++) {
    float v = B2[j];
    for (int k = 0; k < 64; k++) v += h1[k] * W2[k * 64 + j];
    h2[j] = v > 0.f ? v : 0.f;
  }
  for (int j = 0; j < 64; j++) {
    float v = B3[j];
    for (int k = 0; k < 64; k++) v += h2[k] * W3[k * 64 + j];
    h1[j] = v > 0.f ? v : 0.f;
  }
  float z0 = B4[0], z1 = B4[1];
  for (int k = 0; k < 64; k++) { z0 += h1[k] * W4[k * 2]; z1 += h1[k] * W4[k * 2 + 1]; }
  float mx = z0 > z1 ? z0 : z1;
  float lse = mx + logf(expf(z0 - mx) + expf(z1 - mx));
  out[gr * 2 + 0] = z0 - lse;
  out[gr * 2 + 1] = z1 - lse;
}

// ---------------- launch ----------------
extern "C" void kernel_launch(void* const* d_in, const int* in_sizes, int n_in,
                              void* d_out, int out_size, void* d_ws, size_t ws_size,
                              hipStream_t stream) {
  (void)in_sizes; (void)n_in; (void)out_size; (void)ws_size;
  const float* x = (const float*)d_in[0];
  const int* src = (const int*)d_in[1];
  const int* dst = src + N_EDGES;
  const float* ea = (const float*)d_in[2];
  const int* batch = (const int*)d_in[3];
  auto F = [&](int i) { return (const float*)d_in[i]; };
  // param order: conv1..4 {be_w1,be_b1,be_w2,be_b2,mlp_w1,mlp_b1,mlp_w2,mlp_b2,eps} @4..39,
  // bn{1..4}_{g,b} @40..47, fc{1..4}_{w,b} @48..55

  const size_t NM = (size_t)N_NODES * 64;  // 3.2M
  float* ws = (float*)d_ws;
  float* x1 = ws;
  float* x2 = x1 + NM;
  float* x3 = x2 + NM;
  float* x4 = x3 + NM;
  float* y = x4 + NM;
  float* agg = y + NM;
  float* sums = agg + NM;
  float* sumsq = sums + 64;
  float* scale = sumsq + 64;
  float* shift = scale + 64;
  float* pool = shift + 64;      // [64][256]
  float* cnt = pool + 64 * 256;  // [64]
  float* gbuf = cnt + 64;        // [64][256]

  // ---- layer 1 (scalar, dim1=6) ----
  zero_f32<<<256, 256, 0, stream>>>(agg, (int)NM);
  zero_f32<<<1, 256, 0, stream>>>(sums, 128);
  edge_conv1<<<N_EDGES / 256, 256, 0, stream>>>(x, ea, src, dst, F(4), F(5), F(6), F(7), agg);
  node_conv1<<<(N_NODES + 127) / 128, 128, 0, stream>>>(x, agg, F(8), F(9), F(10), F(11), F(12), y);
  col_stats<<<98, dim3(64, 8), 0, stream>>>(y, sums, sumsq);
  bn_finalize<<<1, 64, 0, stream>>>(sums, sumsq, F(40), F(41), scale, shift);
  bn_apply<<<12500, 256, 0, stream>>>(y, scale, shift, x1);

  // ---- layers 2-4 (WMMA) ----
  const float* xprev = x1;
  float* xl[3] = {x2, x3, x4};
  for (int l = 0; l < 3; l++) {
    int pb = 13 + 9 * l;
    zero_f32<<<256, 256, 0, stream>>>(agg, (int)NM);
    zero_f32<<<1, 256, 0, stream>>>(sums, 128);
    edge_wmma<<<N_EDGES / 64, 128, 0, stream>>>(xprev, ea, src, dst,
                                                F(pb), F(pb + 1), F(pb + 2), F(pb + 3), agg);
    node_wmma<<<(N_NODES / 16 + 3) / 4, 128, 0, stream>>>(xprev, agg, F(pb + 4), F(pb + 5),
                                                          F(pb + 6), F(pb + 7), F(pb + 8), y);
    col_stats<<<98, dim3(64, 8), 0, stream>>>(y, sums, sumsq);
    bn_finalize<<<1, 64, 0, stream>>>(sums, sumsq, F(42 + 2 * l), F(43 + 2 * l), scale, shift);
    bn_apply<<<12500, 256, 0, stream>>>(y, scale, shift, xl[l]);
    xprev = xl[l];
  }

  // ---- pooling + head ----
  zero_f32<<<65, 256, 0, stream>>>(pool, 64 * 256 + 64);  // pool + cnt (contiguous)
  count_nodes<<<(N_NODES + 255) / 256, 256, 0, stream>>>(batch, cnt);
  pool_sum<<<12500, 256, 0, stream>>>(x1, x2, x3, x4, batch, pool);
  pool_mean<<<64, 256, 0, stream>>>(pool, cnt, gbuf);
  head<<<1, 64, 0, stream>>>(gbuf, F(48), F(49), F(50), F(51), F(52), F(53), F(54), F(55),
                             (float*)d_out);
}